// GCN_49254684950634
// MI455X (gfx1250) — compile-verified
//
#include <hip/hip_runtime.h>
#include <math.h>

typedef float v2f __attribute__((ext_vector_type(2)));
typedef float v8f __attribute__((ext_vector_type(8)));

#define NN 50000
#define NE 800000
#define DD 64
#define NG 1024
#define NL 4
#define BN_EPS 1e-5f

// ---------------------------------------------------------------------------
// Generic zero-fill (grid-stride)
__global__ void gcn_zero_kernel(float* __restrict__ p, long n) {
    long i = (long)blockIdx.x * blockDim.x + threadIdx.x;
    long stride = (long)gridDim.x * blockDim.x;
    for (; i < n; i += stride) p[i] = 0.0f;
}

// ---------------------------------------------------------------------------
// deg[dst] += 1 per edge
__global__ void gcn_degree_kernel(const int* __restrict__ dst, float* __restrict__ deg, int nE) {
    int i = blockIdx.x * blockDim.x + threadIdx.x;
    if (i < nE) atomicAdd(&deg[dst[i]], 1.0f);
}

// deg -> dinv = rsqrt(deg + 1) in place
__global__ void gcn_dinv_kernel(float* __restrict__ deg, int n) {
    int i = blockIdx.x * blockDim.x + threadIdx.x;
    if (i < n) deg[i] = rsqrtf(deg[i] + 1.0f);
}

// ---------------------------------------------------------------------------
// out[M,64] = A[M,64] @ W[64,64] (+ bias, relu if fused) using fp32 WMMA.
// One wave handles a 16-row tile, producing 4 N-tiles of 16x16.
// K=64 consumed as 16 steps of V_WMMA_F32_16X16X4_F32.
__global__ void gcn_gemm64_wmma(const float* __restrict__ A, const float* __restrict__ W,
                                const float* __restrict__ bias, float* __restrict__ out,
                                int M, int fuse_bias_relu) {
    __shared__ float Wl[DD * DD];
    int tid = threadIdx.x;
    for (int i = tid; i < DD * DD; i += blockDim.x) Wl[i] = W[i];
    __syncthreads();

    int wave = tid >> 5;                 // 8 waves / block
    int lane = tid & 31;
    int tile = blockIdx.x * 8 + wave;    // wave-uniform
    if (tile * 16 >= M) return;          // whole wave exits -> EXEC stays all-ones
    int m0 = tile * 16;

    int half = lane >> 4;                // 0: K pair {0,1}, 1: K pair {2,3}
    int l15  = lane & 15;
    int koff = half * 2;

    v8f acc[4];
    #pragma unroll
    for (int n = 0; n < 4; ++n)
        #pragma unroll
        for (int j = 0; j < 8; ++j) acc[n][j] = 0.0f;

    const float* arow = A + (long)(m0 + l15) * DD;
    #pragma unroll 4
    for (int kt = 0; kt < 16; ++kt) {
        int k = kt * 4 + koff;
        v2f a;
        a[0] = arow[k];
        a[1] = arow[k + 1];
        #pragma unroll
        for (int n = 0; n < 4; ++n) {
            v2f b;
            b[0] = Wl[k * DD + n * 16 + l15];
            b[1] = Wl[(k + 1) * DD + n * 16 + l15];
            acc[n] = __builtin_amdgcn_wmma_f32_16x16x4_f32(
                false, a, false, b, (short)0, acc[n], false, false);
        }
    }

    // C/D layout: VGPR r, lanes 0-15 -> (M=r, N=lane), lanes 16-31 -> (M=r+8, N=lane-16)
    int rowoff = half * 8;
    #pragma unroll
    for (int n = 0; n < 4; ++n) {
        int col = n * 16 + l15;
        float bv = fuse_bias_relu ? bias[col] : 0.0f;
        #pragma unroll
        for (int r = 0; r < 8; ++r) {
            float v = acc[n][r];
            if (fuse_bias_relu) { v += bv; v = v > 0.0f ? v : 0.0f; }
            out[(long)(m0 + rowoff + r) * DD + col] = v;
        }
    }
}

// ---------------------------------------------------------------------------
// agg[dst] += hw[src] * dinv[src]*dinv[dst]; one wave per edge, float2 per lane
__global__ void gcn_edge_scatter(const float* __restrict__ hw, const int* __restrict__ src,
                                 const int* __restrict__ dst, const float* __restrict__ dinv,
                                 float* __restrict__ agg, int nE) {
    int gid = blockIdx.x * blockDim.x + threadIdx.x;
    int e = gid >> 5;
    int lane = gid & 31;
    if (e >= nE) return;
    int s = src[e], d = dst[e];
    float norm = dinv[s] * dinv[d];
    const float2* hs = (const float2*)(hw + (long)s * DD);
    float* ad = agg + (long)d * DD;
    float2 v = hs[lane];
    atomicAdd(&ad[lane * 2],     v.x * norm);
    atomicAdd(&ad[lane * 2 + 1], v.y * norm);
}

// ---------------------------------------------------------------------------
// pre = agg + hw*dinv^2 + b (in place into agg); accumulate column sum/sumsq.
// Block = 256 threads = 4 row-groups x 64 cols, covers 64 node rows.
__global__ void gcn_finalize_stats(float* __restrict__ agg, const float* __restrict__ hw,
                                   const float* __restrict__ dinv, const float* __restrict__ b,
                                   float* __restrict__ colsum, float* __restrict__ colsumsq,
                                   int nNodes) {
    __shared__ float ssum[256];
    __shared__ float ssq[256];
    int c  = threadIdx.x & 63;
    int rg = threadIdx.x >> 6;       // 0..3
    int row0 = blockIdx.x * 64;
    float bc = b[c];
    float lsum = 0.0f, lsq = 0.0f;
    for (int r = rg; r < 64; r += 4) {
        int node = row0 + r;
        if (node >= nNodes) break;
        float di = dinv[node];
        long idx = (long)node * DD + c;
        float pre = agg[idx] + hw[idx] * di * di + bc;
        agg[idx] = pre;
        lsum += pre;
        lsq  += pre * pre;
    }
    ssum[threadIdx.x] = lsum;
    ssq[threadIdx.x]  = lsq;
    __syncthreads();
    if (rg == 0) {
        float s = ssum[c] + ssum[c + 64] + ssum[c + 128] + ssum[c + 192];
        float q = ssq[c]  + ssq[c + 64]  + ssq[c + 128]  + ssq[c + 192];
        atomicAdd(&colsum[c], s);
        atomicAdd(&colsumsq[c], q);
    }
}

// scale/shift from accumulated stats; 64 threads
__global__ void gcn_bn_params(const float* __restrict__ stats, const float* __restrict__ gamma,
                              const float* __restrict__ beta, float* __restrict__ bnp, float invN) {
    int c = threadIdx.x;
    float mean  = stats[c] * invN;
    float var   = stats[64 + c] * invN - mean * mean;
    float scale = gamma[c] * rsqrtf(var + BN_EPS);
    bnp[c]      = scale;
    bnp[64 + c] = beta[c] - mean * scale;
}

// h_out = relu(pre * scale + shift)
__global__ void gcn_bn_apply(const float* __restrict__ pre, const float* __restrict__ bnp,
                             float* __restrict__ out, long n) {
    long i = (long)blockIdx.x * blockDim.x + threadIdx.x;
    if (i >= n) return;
    int c = (int)(i & 63);
    float v = pre[i] * bnp[c] + bnp[64 + c];
    out[i] = v > 0.0f ? v : 0.0f;
}

// pooled[batch[node]] += h[node]
__global__ void gcn_pool(const float* __restrict__ h, const int* __restrict__ batch,
                         float* __restrict__ pooled, int nNodes) {
    long i = (long)blockIdx.x * blockDim.x + threadIdx.x;
    if (i >= (long)nNodes * DD) return;
    int node = (int)(i >> 6);
    int c = (int)(i & 63);
    atomicAdd(&pooled[(long)batch[node] * DD + c], h[i]);
}

// out[g] = dot(z[g,:64], w3) + b3 ; one wave per graph
__global__ void gcn_final_dot(const float* __restrict__ z, const float* __restrict__ w3,
                              const float* __restrict__ b3, float* __restrict__ out, int G) {
    int gid = blockIdx.x * blockDim.x + threadIdx.x;
    int g = gid >> 5;
    int lane = gid & 31;
    if (g >= G) return;
    const float2* zr = (const float2*)(z + (long)g * DD);
    const float2* wr = (const float2*)w3;
    float2 zv = zr[lane];
    float2 wv = wr[lane];
    float p = zv.x * wv.x + zv.y * wv.y;
    #pragma unroll
    for (int off = 16; off > 0; off >>= 1) p += __shfl_down(p, off, 32);
    if (lane == 0) out[g] = p + b3[0];
}

// ---------------------------------------------------------------------------
extern "C" void kernel_launch(void* const* d_in, const int* in_sizes, int n_in,
                              void* d_out, int out_size, void* d_ws, size_t ws_size,
                              hipStream_t stream) {
    const float* x      = (const float*)d_in[0];
    const int*   ei     = (const int*)d_in[1];
    const int*   batch  = (const int*)d_in[2];
    const float* Ws     = (const float*)d_in[3];
    const float* bs     = (const float*)d_in[4];
    const float* gamma  = (const float*)d_in[5];
    const float* beta   = (const float*)d_in[6];
    const float* w1     = (const float*)d_in[7];
    const float* b1     = (const float*)d_in[8];
    const float* w2     = (const float*)d_in[9];
    const float* b2     = (const float*)d_in[10];
    const float* w3     = (const float*)d_in[11];
    const float* b3     = (const float*)d_in[12];
    float* out = (float*)d_out;

    const int* src = ei;
    const int* dst = ei + NE;

    // workspace layout (floats)
    float* ws      = (float*)d_ws;
    float* dinv    = ws;                       // 50048 (padded)
    float* h0      = dinv + 50048;             // NN*DD
    float* hw      = h0 + (long)NN * DD;       // NN*DD
    float* agg     = hw + (long)NN * DD;       // NN*DD (also "pre")
    float* h1      = agg + (long)NN * DD;      // NN*DD
    float* stats   = h1 + (long)NN * DD;       // 128 (colsum|colsumsq)
    float* bnp     = stats + 128;              // 128 (scale|shift)
    float* pooled  = bnp + 128;                // NG*DD
    float* z1      = pooled + (long)NG * DD;   // NG*DD
    float* z2      = z1 + (long)NG * DD;       // NG*DD

    const long NND = (long)NN * DD;

    // degree -> dinv
    gcn_zero_kernel<<<256, 256, 0, stream>>>(dinv, NN);
    gcn_degree_kernel<<<(NE + 255) / 256, 256, 0, stream>>>(dst, dinv, NE);
    gcn_dinv_kernel<<<(NN + 255) / 256, 256, 0, stream>>>(dinv, NN);

    const float* h_in = x;
    float* h_bufs[2] = {h0, h1};
    for (int l = 0; l < NL; ++l) {
        float* h_out = h_bufs[l & 1];
        // hw = h_in @ W[l]
        gcn_gemm64_wmma<<<(NN / 16 + 7) / 8, 256, 0, stream>>>(
            h_in, Ws + (long)l * DD * DD, nullptr, hw, NN, 0);
        // agg = scatter(hw * norm)
        gcn_zero_kernel<<<4096, 256, 0, stream>>>(agg, NND);
        gcn_zero_kernel<<<1, 128, 0, stream>>>(stats, 128);
        gcn_edge_scatter<<<(NE * 32 + 255) / 256, 256, 0, stream>>>(hw, src, dst, dinv, agg, NE);
        // pre = agg + hw*dinv^2 + b ; column stats
        gcn_finalize_stats<<<(NN + 63) / 64, 256, 0, stream>>>(
            agg, hw, dinv, bs + (long)l * DD, stats, stats + 64, NN);
        gcn_bn_params<<<1, 64, 0, stream>>>(stats, gamma + (long)l * DD, beta + (long)l * DD,
                                            bnp, 1.0f / (float)NN);
        gcn_bn_apply<<<(int)((NND + 255) / 256), 256, 0, stream>>>(agg, bnp, h_out, NND);
        h_in = h_out;
    }

    // global add pool
    gcn_zero_kernel<<<256, 256, 0, stream>>>(pooled, (long)NG * DD);
    gcn_pool<<<(int)((NND + 255) / 256), 256, 0, stream>>>(h_in, batch, pooled, NN);

    // MLP: two fused bias+relu WMMA GEMMs, then [64,1] projection
    gcn_gemm64_wmma<<<(NG / 16 + 7) / 8, 256, 0, stream>>>(pooled, w1, b1, z1, NG, 1);
    gcn_gemm64_wmma<<<(NG / 16 + 7) / 8, 256, 0, stream>>>(z1, w2, b2, z2, NG, 1);
    gcn_final_dot<<<(NG * 32 + 255) / 256, 256, 0, stream>>>(z2, w3, b3, out, NG);
}